// Encoder_SPLIT_intermediate_36292473652060
// MI455X (gfx1250) — compile-verified
//
#include <hip/hip_runtime.h>
#include <hip/hip_bf16.h>

// MI455X / gfx1250 implementation.
// Phase 1: tiled transpose of x for coalesced LSTM reads.
// Phase 2: per-thread 2-layer LSTM (H=4) -> ys[path][B][T*4]  (f32),
//          using native v_tanh_f32 when available.
// Phase 3: WMMA f32 16x16x4 GEMM  ys[B,2048] @ W^T[2048,128] + b, tanh, concat.
//          B operand staged in LDS as contiguous K-pairs (direct ds_load_b64
//          into WMMA-ready even pairs), staged via async load-to-LDS.

#define BATCH   8192
#define SEQ     512
#define TC      1024          // SEQ*2 channels interleaved
#define KDIM    2048          // SEQ*4
#define NOUT    128
#define OUTC    256
#define BPITCH  290           // dwords per (kk,half) block: 256 data + pad;
                              // 290 % 64 == 34 -> staging stores conflict-free

typedef __attribute__((ext_vector_type(2))) float v2f;
typedef __attribute__((ext_vector_type(8))) float v8f;

#if defined(__HIP_DEVICE_COMPILE__) && __has_builtin(__builtin_amdgcn_global_load_async_to_lds_b32)
#define HAVE_ASYNC_LDS 1
#else
#define HAVE_ASYNC_LDS 0
#endif

typedef __attribute__((address_space(1))) int* gptr_i32;
typedef __attribute__((address_space(3))) int* lptr_i32;

__device__ __forceinline__ float tanh_f(float x) {
#if __has_builtin(__builtin_amdgcn_tanh_f32)
    return __builtin_amdgcn_tanh_f32(x);        // native v_tanh_f32 (TRANS32)
#else
    x = fminf(fmaxf(x, -15.0f), 15.0f);         // avoid inf/inf
    float e = __expf(-2.0f * x);
    return (1.0f - e) / (1.0f + e);
#endif
}
__device__ __forceinline__ float sigmoid_f(float x) {
#if __has_builtin(__builtin_amdgcn_tanh_f32)
    return fmaf(0.5f, __builtin_amdgcn_tanh_f32(0.5f * x), 0.5f);
#else
    return 1.0f / (1.0f + __expf(-x));          // x->-inf: exp->inf -> 0, no NaN
#endif
}

__device__ __forceinline__ void wait_async_zero() {
#if __has_builtin(__builtin_amdgcn_s_wait_asynccnt)
    __builtin_amdgcn_s_wait_asynccnt(0);
#else
    asm volatile("s_wait_asynccnt 0" ::: "memory");
#endif
}

// ---------------------------------------------------------------------------
// Kernel 1: transpose x [8192 x 1024] -> xT [1024 x 8192] via LDS tile.
// ---------------------------------------------------------------------------
__global__ __launch_bounds__(256) void k_transpose_x(
    const float* __restrict__ x, float* __restrict__ xT)
{
    __shared__ float tile[32][33];
    const int tx = threadIdx.x;           // 0..31
    const int ty = threadIdx.y;           // 0..7
    const int col0 = blockIdx.x * 32;     // over 1024
    const int row0 = blockIdx.y * 32;     // over 8192
#pragma unroll
    for (int r = 0; r < 32; r += 8)
        tile[ty + r][tx] = x[(size_t)(row0 + ty + r) * TC + (col0 + tx)];
    __syncthreads();
#pragma unroll
    for (int r = 0; r < 32; r += 8)
        xT[(size_t)(col0 + ty + r) * BATCH + (row0 + tx)] = tile[tx][ty + r];
}

// ---------------------------------------------------------------------------
// Kernel 2: fused 2-layer LSTM. One thread = one (path, batch element).
// xT rows are (2*t + path); lane = batch -> fully coalesced reads.
// Output ys[path][b][t*4+g] == reference reshape(B, 4*T) layout.
// ---------------------------------------------------------------------------
__global__ __launch_bounds__(256) void k_lstm2(
    const float* __restrict__ xT,     // [1024][8192]
    float* __restrict__ ys,           // [2][8192][2048]
    const float* __restrict__ hWih0, const float* __restrict__ hWhh0,
    const float* __restrict__ hbih0, const float* __restrict__ hbhh0,
    const float* __restrict__ hWih1, const float* __restrict__ hWhh1,
    const float* __restrict__ hbih1, const float* __restrict__ hbhh1,
    const float* __restrict__ lWih0, const float* __restrict__ lWhh0,
    const float* __restrict__ lbih0, const float* __restrict__ lbhh0,
    const float* __restrict__ lWih1, const float* __restrict__ lWhh1,
    const float* __restrict__ lbih1, const float* __restrict__ lbhh1)
{
    const int p = blockIdx.y;                               // 0 = H, 1 = L
    const int b = blockIdx.x * blockDim.x + threadIdx.x;    // batch element

    const float* Wih0 = p ? lWih0 : hWih0;
    const float* Whh0 = p ? lWhh0 : hWhh0;
    const float* Bi0  = p ? lbih0 : hbih0;
    const float* Bh0  = p ? lbhh0 : hbhh0;
    const float* Wih1 = p ? lWih1 : hWih1;
    const float* Whh1 = p ? lWhh1 : hWhh1;
    const float* Bi1  = p ? lbih1 : hbih1;
    const float* Bh1  = p ? lbhh1 : hbhh1;

    // Wave-uniform weight loads -> scalar loads / registers.
    float wih0[16], bb0[16], bb1[16];
    float whh0[64], wih1[64], whh1[64];
#pragma unroll
    for (int j = 0; j < 16; ++j) {
        wih0[j] = Wih0[j];
        bb0[j]  = Bi0[j] + Bh0[j];
        bb1[j]  = Bi1[j] + Bh1[j];
    }
#pragma unroll
    for (int j = 0; j < 64; ++j) {
        whh0[j] = Whh0[j];
        wih1[j] = Wih1[j];
        whh1[j] = Whh1[j];
    }

    float h0[4] = {0,0,0,0}, c0[4] = {0,0,0,0};
    float h1[4] = {0,0,0,0}, c1[4] = {0,0,0,0};

    float4* yout = (float4*)(ys + ((size_t)p * BATCH + b) * KDIM);

    for (int t = 0; t < SEQ; ++t) {
        const float xt = xT[(size_t)(2 * t + p) * BATCH + b];
        float g[16];
        // ---- layer 0: g = xt*Wih0 + h0*Whh0^T + b ----
#pragma unroll
        for (int j = 0; j < 16; ++j) {
            float a = fmaf(wih0[j], xt, bb0[j]);
            a = fmaf(whh0[j*4+0], h0[0], a);
            a = fmaf(whh0[j*4+1], h0[1], a);
            a = fmaf(whh0[j*4+2], h0[2], a);
            a = fmaf(whh0[j*4+3], h0[3], a);
            g[j] = a;
        }
#pragma unroll
        for (int k = 0; k < 4; ++k) {   // gate order: i,f,g,o
            c0[k] = sigmoid_f(g[4+k]) * c0[k] + sigmoid_f(g[k]) * tanh_f(g[8+k]);
            h0[k] = sigmoid_f(g[12+k]) * tanh_f(c0[k]);
        }
        // ---- layer 1: g = h0*Wih1^T + h1*Whh1^T + b ----
#pragma unroll
        for (int j = 0; j < 16; ++j) {
            float a = bb1[j];
            a = fmaf(wih1[j*4+0], h0[0], a);
            a = fmaf(wih1[j*4+1], h0[1], a);
            a = fmaf(wih1[j*4+2], h0[2], a);
            a = fmaf(wih1[j*4+3], h0[3], a);
            a = fmaf(whh1[j*4+0], h1[0], a);
            a = fmaf(whh1[j*4+1], h1[1], a);
            a = fmaf(whh1[j*4+2], h1[2], a);
            a = fmaf(whh1[j*4+3], h1[3], a);
            g[j] = a;
        }
#pragma unroll
        for (int k = 0; k < 4; ++k) {
            c1[k] = sigmoid_f(g[4+k]) * c1[k] + sigmoid_f(g[k]) * tanh_f(g[8+k]);
            h1[k] = sigmoid_f(g[12+k]) * tanh_f(c1[k]);
        }
        yout[t] = make_float4(h1[0], h1[1], h1[2], h1[3]);
    }
}

// ---------------------------------------------------------------------------
// Kernel 3: out[:, off:off+128] = tanh(ys @ W^T + bias) using
// V_WMMA_F32_16X16X4_F32.  Block = 128 threads (4 waves); wave w owns batch
// rows b0..b0+15 and all 128 output columns (8 accumulator tiles).
//
// ISA layouts (CDNA5 ISA §7.12.2), wave32:
//   A 16x4 f32 : lane%16 = M row, K = vgpr + 2*(lane>=16)
//   B 4x16 f32 : lane%16 = N col, K = vgpr + 2*(lane>=16)   (mirror of A)
//   C/D 16x16  : lane%16 = N col, M row = vgpr + 8*(lane>=16)
//
// LDS stores each WMMA K-pair contiguously:
//   Bs[(k>>1)*BPITCH + n*2 + (k&1)]  with  k = K-k0
// so the B operand is a single aligned ds_load_b64 into an even VGPR pair.
// ---------------------------------------------------------------------------
__global__ __launch_bounds__(128) void k_gemm_tanh(
    const float* __restrict__ ysH,    // [8192][2048]
    const float* __restrict__ ysL,
    const float* __restrict__ WH,     // [128][2048]
    const float* __restrict__ WL,
    const float* __restrict__ bH,     // [128]
    const float* __restrict__ bL,
    float* __restrict__ out)          // [8192][256]
{
    __shared__ __align__(16) float Bs[32 * BPITCH];   // 37,120 B

    const int p    = blockIdx.y;
    const float* ysM  = p ? ysL : ysH;
    const float* W    = p ? WL  : WH;
    const float* bias = p ? bL  : bH;
    const int colOff  = p * NOUT;

    const int tid  = threadIdx.x;
    const int lane = tid & 31;
    const int wave = tid >> 5;
    const int half = lane >> 4;                 // 0 | 1
    const int lm   = lane & 15;
    const int b0   = blockIdx.x * 64 + wave * 16;

    v8f acc[8];
#pragma unroll
    for (int nt = 0; nt < 8; ++nt)
        acc[nt] = (v8f){0.f,0.f,0.f,0.f,0.f,0.f,0.f,0.f};

    const float* aRow = ysM + (size_t)(b0 + lm) * KDIM;

    for (int kb = 0; kb < KDIM / 64; ++kb) {
        const int k0 = kb * 64;
        // Stage W^T chunk: Bs[(k>>1)*BPITCH + n*2 + (k&1)] = W[n][k0+k].
        // Global reads coalesced over k; LDS writes bank-conflict-free
        // (stride 290 dwords == 34 mod 64).
        for (int i = tid; i < 64 * NOUT; i += 128) {
            const int n = i >> 6;
            const int k = i & 63;
            const int ldsIdx = (k >> 1) * BPITCH + n * 2 + (k & 1);
            const float* gsrc = W + (size_t)n * KDIM + k0 + k;
#if HAVE_ASYNC_LDS
            __builtin_amdgcn_global_load_async_to_lds_b32(
                (gptr_i32)(size_t)gsrc,
                (lptr_i32)&Bs[ldsIdx],
                0, 0);
#else
            Bs[ldsIdx] = *gsrc;
#endif
        }
#if HAVE_ASYNC_LDS
        wait_async_zero();
#endif
        __syncthreads();

        if (kb + 1 < KDIM / 64)
            __builtin_prefetch(aRow + k0 + 64, 0, 1);   // global_prefetch next A chunk

#pragma unroll
        for (int kk = 0; kk < 16; ++kk) {
            const int kIdx = k0 + kk * 4 + 2 * half;
            const v2f a = *(const v2f*)(aRow + kIdx);                  // b64, 8B aligned
            const float* bBase = &Bs[(kk * 2 + half) * BPITCH + lm * 2];
#pragma unroll
            for (int nt = 0; nt < 8; ++nt) {
                const v2f bv = *(const v2f*)(bBase + nt * 32);         // ds_load_b64
                acc[nt] = __builtin_amdgcn_wmma_f32_16x16x4_f32(
                    /*neg_a=*/false, a, /*neg_b=*/false, bv,
                    /*c_mod=*/(short)0, acc[nt],
                    /*reuse_a=*/false, /*reuse_b=*/false);
            }
        }
        __syncthreads();
    }

    // Epilogue: bias + tanh + store (C layout -> row-major out).
#pragma unroll
    for (int nt = 0; nt < 8; ++nt) {
        const int n = nt * 16 + lm;
        const float bb = bias[n];
#pragma unroll
        for (int j = 0; j < 8; ++j) {
            const int row = b0 + j + 8 * half;
            out[(size_t)row * OUTC + colOff + n] = tanh_f(acc[nt][j] + bb);
        }
    }
}

// ---------------------------------------------------------------------------
extern "C" void kernel_launch(void* const* d_in, const int* in_sizes, int n_in,
                              void* d_out, int out_size, void* d_ws, size_t ws_size,
                              hipStream_t stream)
{
    const float* x      = (const float*)d_in[0];
    const float* hWih0  = (const float*)d_in[1];
    const float* hWhh0  = (const float*)d_in[2];
    const float* hbih0  = (const float*)d_in[3];
    const float* hbhh0  = (const float*)d_in[4];
    const float* hWih1  = (const float*)d_in[5];
    const float* hWhh1  = (const float*)d_in[6];
    const float* hbih1  = (const float*)d_in[7];
    const float* hbhh1  = (const float*)d_in[8];
    const float* lWih0  = (const float*)d_in[9];
    const float* lWhh0  = (const float*)d_in[10];
    const float* lbih0  = (const float*)d_in[11];
    const float* lbhh0  = (const float*)d_in[12];
    const float* lWih1  = (const float*)d_in[13];
    const float* lWhh1  = (const float*)d_in[14];
    const float* lbih1  = (const float*)d_in[15];
    const float* lbhh1  = (const float*)d_in[16];
    const float* WH     = (const float*)d_in[17];
    const float* bH     = (const float*)d_in[18];
    const float* WL     = (const float*)d_in[19];
    const float* bL     = (const float*)d_in[20];
    float* out = (float*)d_out;

    // Workspace layout: xT (32 MB) | ys[2] (128 MB)
    float* xT  = (float*)d_ws;
    float* ys  = (float*)((char*)d_ws + (size_t)TC * BATCH * sizeof(float));
    float* ysH = ys;
    float* ysL = ys + (size_t)BATCH * KDIM;

    k_transpose_x<<<dim3(TC / 32, BATCH / 32), dim3(32, 8), 0, stream>>>(x, xT);

    k_lstm2<<<dim3(BATCH / 256, 2), 256, 0, stream>>>(
        xT, ys,
        hWih0, hWhh0, hbih0, hbhh0, hWih1, hWhh1, hbih1, hbhh1,
        lWih0, lWhh0, lbih0, lbhh0, lWih1, lWhh1, lbih1, lbhh1);

    k_gemm_tanh<<<dim3(BATCH / 64, 2), 128, 0, stream>>>(
        ysH, ysL, WH, WL, bH, bL, out);
}